// TDTFPredictiveRouter_22488448761976
// MI455X (gfx1250) — compile-verified
//
#include <hip/hip_runtime.h>
#include <hip/hip_bf16.h>
#include <stdint.h>

#define DEV __device__ __forceinline__

// Problem shape (from reference): B=8, T=8192, D=2048, fp32.
constexpr int Bb   = 8;
constexpr int Tt   = 8192;
constexpr int Dk   = 2048;
constexpr int ROWS = Bb * Tt;     // 65536
constexpr int TOPK = Tt / 4;      // 2048 (= max(1, int(T*0.25)))

// ---- Kernel-1 tiling (wave = 16 rows, staged through LDS) ----
constexpr int WROWS   = 16;              // rows per wave (WMMA M)
constexpr int NWAVES  = 4;               // waves per workgroup
constexpr int TILE_K  = 32;              // K columns staged per tile
constexpr int RS      = 36;              // padded LDS row stride (floats): 36 mod 64 -> conflict-free b64 reads
constexpr int NT      = Dk / TILE_K;     // 64 tiles per row-set
constexpr int TILE_FLOATS = WROWS * RS;          // one operand tile
constexpr int BUF_FLOATS  = 2 * TILE_FLOATS;     // a-tile + p-tile
constexpr int WAVE_FLOATS = 2 * BUF_FLOATS;      // double buffered
constexpr int SMEM_FLOATS = NWAVES * WAVE_FLOATS; // 9216 floats = 36 KB

typedef float v2f __attribute__((ext_vector_type(2)));
typedef float v8f __attribute__((ext_vector_type(8)));

#if __has_builtin(__builtin_amdgcn_wmma_f32_16x16x4_f32)
#define USE_WMMA_F32 1
#else
#define USE_WMMA_F32 0
#endif

// ---------- CDNA5 async global->LDS (ASYNCcnt-tracked) ----------
DEV void async_b128_to_lds(unsigned lds_addr, const float* gptr) {
  asm volatile("global_load_async_to_lds_b128 %0, %1, off"
               :: "v"(lds_addr), "v"((unsigned long long)(uintptr_t)gptr)
               : "memory");
}
DEV void wait_async8() { asm volatile("s_wait_asynccnt 8" ::: "memory"); }
DEV void wait_async0() { asm volatile("s_wait_asynccnt 0" ::: "memory"); }

// ---------------------------------------------------------------
// K1: per-row D_st = sum(a^2)/D, D_ch = sum((a-p)^2)/D
// One wave per 16 rows. Tiles 16x32 fp32 staged via async loads,
// reduced with V_WMMA_F32_16X16X4_F32 against an all-ones B matrix:
// D[m][n] += sum_k A[m][k] * 1  -> every column of C holds row sums.
// ---------------------------------------------------------------
__global__ __launch_bounds__(NWAVES * 32)
void surprise_kernel(const float* __restrict__ A, const float* __restrict__ P,
                     float* __restrict__ Dst, float* __restrict__ Dch)
{
  __shared__ float smem[SMEM_FLOATS];
  const int lane = threadIdx.x & 31;
  const int wave = threadIdx.x >> 5;
  const int row_base = blockIdx.x * (NWAVES * WROWS) + wave * WROWS;
  float* wbase = smem + wave * WAVE_FLOATS;
  const float* gA = A + (size_t)row_base * Dk;
  const float* gP = P + (size_t)row_base * Dk;

  // Issue one 16x32 tile (both operands) = 8 async b128 instructions.
  auto issue = [&](int t, int buf) {
    const int k0 = t * TILE_K;
    float* bb = wbase + buf * BUF_FLOATS;
    for (int it = 0; it < 4; ++it) {
      int flat = it * 32 + lane;
      int r = flat >> 3;            // tile row 0..15
      int c = (flat & 7) << 2;      // col 0,4,...,28 (float4 granularity)
      async_b128_to_lds((unsigned)(uintptr_t)(bb + r * RS + c),
                        gA + (size_t)r * Dk + (k0 + c));
      async_b128_to_lds((unsigned)(uintptr_t)(bb + TILE_FLOATS + r * RS + c),
                        gP + (size_t)r * Dk + (k0 + c));
    }
  };

#if USE_WMMA_F32
  v8f c_st = {};
  v8f c_ch = {};
  const v2f ones = {1.0f, 1.0f};
#else
  float s_st = 0.f, s_ch = 0.f;
#endif
  // A-fragment addressing: lane L -> row L&15, K pair 2*(L>>4)
  const int ar  = lane & 15;
  const int akc = (lane >> 4) << 1;

  issue(0, 0);
  for (int t = 0; t < NT; ++t) {
    const int cur = t & 1;
    if (t + 1 < NT) { issue(t + 1, cur ^ 1); wait_async8(); }
    else            { wait_async0(); }

    const float* ta = wbase + cur * BUF_FLOATS + ar * RS + akc;
    const float* tp = ta + TILE_FLOATS;
    for (int kk = 0; kk < TILE_K; kk += 4) {
      v2f av = *(const v2f*)(ta + kk);   // ds_load_b64, bank-conflict-free
      v2f pv = *(const v2f*)(tp + kk);
      v2f a2 = av * av;
      v2f dv = av - pv;
      v2f d2 = dv * dv;
#if USE_WMMA_F32
      c_st = __builtin_amdgcn_wmma_f32_16x16x4_f32(
          false, a2, false, ones, (short)0, c_st, false, false);
      c_ch = __builtin_amdgcn_wmma_f32_16x16x4_f32(
          false, d2, false, ones, (short)0, c_ch, false, false);
#else
      s_st += a2.x + a2.y;
      s_ch += d2.x + d2.y;
#endif
    }
  }

  const float invD = 1.0f / (float)Dk;
#if USE_WMMA_F32
  // C layout: VGPR r -> lanes 0-15 hold M=r (any column), lanes 16-31 hold M=8+r.
  if (lane == 0) {
    for (int r = 0; r < 8; ++r) {
      Dst[row_base + r] = c_st[r] * invD;
      Dch[row_base + r] = c_ch[r] * invD;
    }
  } else if (lane == 16) {
    for (int r = 0; r < 8; ++r) {
      Dst[row_base + 8 + r] = c_st[r] * invD;
      Dch[row_base + 8 + r] = c_ch[r] * invD;
    }
  }
#else
  s_st += __shfl_xor(s_st, 16, 32);
  s_ch += __shfl_xor(s_ch, 16, 32);
  if (lane < 16) {
    Dst[row_base + lane] = s_st * invD;
    Dch[row_base + lane] = s_ch * invD;
  }
#endif
}

// ---------------------------------------------------------------
// K2: deterministic mean of D_st + precompute scalar params
// params: [0]=mean(D_st) [1]=log(o_ce+eps) [2]=softplus(b_ce) [3]=softplus(b_cu)
// ---------------------------------------------------------------
DEV float softplusf(float x) { return fmaxf(x, 0.f) + log1pf(expf(-fabsf(x))); }

__global__ void reduce_kernel(const float* __restrict__ Dst,
                              const float* o_ce, const float* b_ce, const float* b_cu,
                              float* __restrict__ params)
{
  __shared__ float red[256];
  float s = 0.f;
  for (int i = threadIdx.x; i < ROWS; i += 256) s += Dst[i];
  red[threadIdx.x] = s;
  __syncthreads();
  for (int st = 128; st > 0; st >>= 1) {
    if ((int)threadIdx.x < st) red[threadIdx.x] += red[threadIdx.x + st];
    __syncthreads();
  }
  if (threadIdx.x == 0) {
    params[0] = red[0] / (float)ROWS;
    params[1] = logf(*o_ce + 1e-10f);
    params[2] = softplusf(*b_ce);
    params[3] = softplusf(*b_cu);
  }
}

// ---------------------------------------------------------------
// K3: continuous gate g = S_CE + S_CU - S_CE*S_CU
// ---------------------------------------------------------------
__global__ void gate_kernel(const float* __restrict__ Dst, const float* __restrict__ Dch,
                            const float* __restrict__ params, const float* m_cu,
                            float* __restrict__ g)
{
  int i = blockIdx.x * 256 + threadIdx.x;
  if (i >= ROWS) return;
  const float ma = params[0], lo = params[1], bce = params[2], bcu = params[3];
  float dst = Dst[i], dch = Dch[i];
  float CE = dst - (dch - lo);
  float CU = dst - (*m_cu) * ma;
  float sce = 1.f / (1.f + expf(-bce * CE));
  float scu = 1.f / (1.f + expf(-bcu * CU));
  g[i] = sce + scu - sce * scu;
}

// ---------------------------------------------------------------
// Top-k: exact radix-select on order-preserving uint32 keys.
// ---------------------------------------------------------------
DEV unsigned fkey(float f) {
  unsigned u = __float_as_uint(f);
  return (u & 0x80000000u) ? ~u : (u | 0x80000000u);
}

__global__ void select_kernel(const float* __restrict__ g,
                              unsigned* __restrict__ thresh, unsigned* __restrict__ remain)
{
  const int b = blockIdx.x;
  const float* row = g + (size_t)b * Tt;
  __shared__ unsigned hist[256];
  __shared__ unsigned s_prefix;
  __shared__ int s_kneed;
  if (threadIdx.x == 0) { s_prefix = 0u; s_kneed = TOPK; }
  __syncthreads();

  for (int pass = 3; pass >= 0; --pass) {
    hist[threadIdx.x] = 0u;
    __syncthreads();
    const unsigned prefix = s_prefix;
    const unsigned hmask = (pass == 3) ? 0u : (0xFFFFFFFFu << ((pass + 1) * 8));
    const int shift = pass * 8;
    for (int j = threadIdx.x; j < Tt; j += 256) {
      unsigned u = fkey(row[j]);
      if ((u & hmask) == (prefix & hmask))
        atomicAdd(&hist[(u >> shift) & 255u], 1u);
    }
    __syncthreads();
    if (threadIdx.x == 0) {
      unsigned cum = 0;
      int sel = 0;
      for (int v = 255; v >= 0; --v) {
        unsigned h = hist[v];
        if (cum + h >= (unsigned)s_kneed) { sel = v; break; }
        cum += h;
      }
      s_prefix = prefix | ((unsigned)sel << shift);
      s_kneed  = s_kneed - (int)cum;   // rank among keys == chosen bucket so far
    }
    __syncthreads();
  }
  if (threadIdx.x == 0) { thresh[b] = s_prefix; remain[b] = (unsigned)s_kneed; }
}

// K5: write binary mask; ties at threshold go to lowest indices (JAX top_k order).
__global__ void mask_kernel(const float* __restrict__ g,
                            const unsigned* __restrict__ thresh,
                            const unsigned* __restrict__ remain,
                            float* __restrict__ mask)
{
  const int b = blockIdx.x;
  const float* row = g + (size_t)b * Tt;
  float* out = mask + (size_t)b * Tt;
  const unsigned th = thresh[b];
  const unsigned rem = remain[b];
  constexpr int C = Tt / 256;  // 32 contiguous elements per thread (index-ordered)
  const int t = threadIdx.x;
  __shared__ unsigned cnt[256];
  __shared__ unsigned scan[256];

  unsigned local = 0;
  for (int j = 0; j < C; ++j)
    if (fkey(row[t * C + j]) == th) local++;
  cnt[t] = local;
  __syncthreads();
  if (t == 0) {
    unsigned acc = 0;
    for (int i = 0; i < 256; ++i) { scan[i] = acc; acc += cnt[i]; }
  }
  __syncthreads();

  unsigned eqr = scan[t];
  for (int j = 0; j < C; ++j) {
    unsigned u = fkey(row[t * C + j]);
    float o = 0.f;
    if (u > th) o = 1.f;
    else if (u == th) { o = (eqr < rem) ? 1.f : 0.f; eqr++; }
    out[t * C + j] = o;
  }
}

// ---------------------------------------------------------------
extern "C" void kernel_launch(void* const* d_in, const int* in_sizes, int n_in,
                              void* d_out, int out_size, void* d_ws, size_t ws_size,
                              hipStream_t stream)
{
  (void)in_sizes; (void)n_in; (void)out_size; (void)ws_size;
  const float* A    = (const float*)d_in[0];
  const float* P    = (const float*)d_in[1];
  const float* o_ce = (const float*)d_in[2];
  const float* m_cu = (const float*)d_in[3];
  const float* b_ce = (const float*)d_in[4];
  const float* b_cu = (const float*)d_in[5];

  float* g    = (float*)d_out;          // [ROWS]
  float* mask = g + ROWS;               // [ROWS]

  float* Dst    = (float*)d_ws;         // [ROWS]
  float* Dch    = Dst + ROWS;           // [ROWS]
  float* params = Dch + ROWS;           // [8]
  unsigned* thr = (unsigned*)(params + 8);   // [Bb]
  unsigned* rem = thr + 16;                  // [Bb]

  surprise_kernel<<<ROWS / (NWAVES * WROWS), NWAVES * 32, 0, stream>>>(A, P, Dst, Dch);
  reduce_kernel<<<1, 256, 0, stream>>>(Dst, o_ce, b_ce, b_cu, params);
  gate_kernel<<<ROWS / 256, 256, 0, stream>>>(Dst, Dch, params, m_cu, g);
  select_kernel<<<Bb, 256, 0, stream>>>(g, thr, rem);
  mask_kernel<<<Bb, 256, 0, stream>>>(g, thr, rem, mask);
}